// RoiAlign_52123723104378
// MI455X (gfx1250) — compile-verified
//
#include <hip/hip_runtime.h>
#include <stdint.h>

// ---------------------------------------------------------------------------
// RoI Align (FPN, 4 levels) for MI455X / gfx1250.
// Memory-bound gather op: CDNA5 async global->LDS B128 loads with
// double-buffered software pipelining (ASYNCcnt), wave32-sized blocks,
// 128-bit LDS reads and 128-bit global stores. Feature pyramid (~178MB)
// fits the 192MB L2, so default RT temporal hints keep gathers in L2.
// Geometry is computed ONCE per stage (at async-issue time) and lx/ly are
// carried in registers to the compute phase of the next pipeline step.
// ---------------------------------------------------------------------------

#define POOLN 7

namespace {
constexpr int kB = 2;
constexpr int kN = 1000;
constexpr int kC = 256;
constexpr int kCellsPerBox  = POOLN * POOLN;                    // 49
constexpr int kCells        = kB * kN * kCellsPerBox;           // 98000
constexpr int kCellsPerStage  = 4;
constexpr int kStagesPerBlock = 4;
constexpr int kCellsPerBlock  = kCellsPerStage * kStagesPerBlock; // 16
constexpr int kBlocks         = kCells / kCellsPerBlock;          // 6125 (exact)
// LDS: 2 buffers x (4 cells x 4 corners x 256 ch) floats = 32 KB
constexpr int kBufFloats = kCellsPerStage * 4 * kC;               // 4096
}

// s_wait_asynccnt <= n  (ASYNCcnt tracks GLOBAL_LOAD_ASYNC_TO_LDS_*)
#if __has_builtin(__builtin_amdgcn_s_wait_asynccnt)
#define ASYNC_WAIT(n) __builtin_amdgcn_s_wait_asynccnt(n)
#else
#define ASYNC_WAIT(n) asm volatile("s_wait_asynccnt " #n ::: "memory")
#endif

struct Geom {
    const float* f;   // selected level base
    int   off[4];     // element offsets of tl, tr, bl, br (channel 0)
    float lx, ly;
};

__device__ __forceinline__ Geom cell_geom(
    int cell, const float* __restrict__ props,
    const float* __restrict__ f0, const float* __restrict__ f1,
    const float* __restrict__ f2, const float* __restrict__ f3)
{
    Geom g;
    int b  = cell / (kN * kCellsPerBox);
    int r  = cell - b * (kN * kCellsPerBox);
    int n  = r / kCellsPerBox;
    int m  = r - n * kCellsPerBox;
    int py = m / POOLN;
    int px = m - py * POOLN;

    const float* p = props + (size_t)(b * kN + n) * 4;
    float x1 = p[0], y1 = p[1], x2 = p[2], y2 = p[3];

    // level select: floor(log2(sqrt(area)/canon)), nan/-inf -> 0, clip [0,3]
    float area  = (y2 - y1) * (x2 - x1);
    const float canon = 56.0f / 1024.0f + 1e-6f;
    float lf = floorf(logf(sqrtf(area) / canon) / logf(2.0f));
    if (!(lf >= 0.0f)) lf = 0.0f;            // catches NaN, -inf, negatives
    int lvl = (lf > 3.0f) ? 3 : (int)lf;

    int H = 256 >> lvl;                       // H == W per level
    g.f = (lvl == 0) ? f0 : (lvl == 1) ? f1 : (lvl == 2) ? f2 : f3;

    float Hf = (float)H;
    float iy = y1 * (Hf - 1.0f) + (float)py * ((y2 - y1) * (Hf - 1.0f) / 6.0f);
    float ix = x1 * (Hf - 1.0f) + (float)px * ((x2 - x1) * (Hf - 1.0f) / 6.0f);
    float tyf = floorf(iy), txf = floorf(ix);
    g.ly = iy - tyf;
    g.lx = ix - txf;

    int ty = (int)tyf, tx = (int)txf;
    int ty_i = min(max(ty,     0), H - 1);
    int by_i = min(max(ty + 1, 0), H - 1);
    int tx_i = min(max(tx,     0), H - 1);
    int bx_i = min(max(tx + 1, 0), H - 1);

    int rowT = (b * H + ty_i) * H;
    int rowB = (b * H + by_i) * H;
    g.off[0] = (rowT + tx_i) * kC;   // tl
    g.off[1] = (rowT + bx_i) * kC;   // tr
    g.off[2] = (rowB + tx_i) * kC;   // bl
    g.off[3] = (rowB + bx_i) * kC;   // br
    return g;
}

__device__ __forceinline__ float bil(float tl, float tr, float bl, float br,
                                     float lx, float ly)
{
    float top = tl + (tr - tl) * lx;
    float bot = bl + (br - bl) * lx;
    return top + (bot - top) * ly;
}

__global__ __launch_bounds__(256) void roi_align_async_kernel(
    const float* __restrict__ f0, const float* __restrict__ f1,
    const float* __restrict__ f2, const float* __restrict__ f3,
    const float* __restrict__ props, float* __restrict__ out)
{
    __shared__ __align__(16) float smem[2 * kBufFloats];   // 32 KB

    const int t     = threadIdx.x;
    const int w     = t & 63;        // channel group: channels 4w..4w+3
    const int cq    = t >> 6;        // cell within stage (0..3)
    const int cell0 = blockIdx.x * kCellsPerBlock;

    // Issue one stage's corner gathers from a precomputed Geom:
    // 4 cells x 4 corners x 256 ch = 16 KB per stage,
    // one GLOBAL_LOAD_ASYNC_TO_LDS_B128 per (thread, corner).
    auto issue_async = [&](const Geom& g, int buf) {
#pragma unroll
        for (int k = 0; k < 4; ++k) {
            const float* src = g.f + g.off[k] + w * 4;        // 16B aligned
            uint32_t lds = (uint32_t)(uintptr_t)
                &smem[buf * kBufFloats + cq * 1024 + k * 256 + w * 4];
            uint64_t ga = (uint64_t)(uintptr_t)src;
            asm volatile("global_load_async_to_lds_b128 %0, %1, off"
                         :: "v"(lds), "v"(ga)
                         : "memory");
        }
    };

    // Prologue: geometry + async issue for stage 0.
    Geom g_cur = cell_geom(cell0 + cq, props, f0, f1, f2, f3);
    issue_async(g_cur, 0);

#pragma unroll
    for (int s = 0; s < kStagesPerBlock; ++s) {
        const float lx = g_cur.lx;           // carried from issue time
        const float ly = g_cur.ly;

        if (s + 1 < kStagesPerBlock) {
            // Prefetch next stage: geometry computed exactly once here.
            Geom g_next = cell_geom(cell0 + (s + 1) * kCellsPerStage + cq,
                                    props, f0, f1, f2, f3);
            issue_async(g_next, (s + 1) & 1);
            g_cur = g_next;
            ASYNC_WAIT(4);                   // stage s resident (next 4 in flight)
        } else {
            ASYNC_WAIT(0);                   // last stage: drain
        }
        __syncthreads();                     // all waves' data visible in LDS

        const int base = (s & 1) * kBufFloats + cq * 1024 + w * 4;
        float4 tl = *(const float4*)&smem[base + 0 * 256];
        float4 tr = *(const float4*)&smem[base + 1 * 256];
        float4 bl = *(const float4*)&smem[base + 2 * 256];
        float4 br = *(const float4*)&smem[base + 3 * 256];

        float4 o;
        o.x = bil(tl.x, tr.x, bl.x, br.x, lx, ly);
        o.y = bil(tl.y, tr.y, bl.y, br.y, lx, ly);
        o.z = bil(tl.z, tr.z, bl.z, br.z, lx, ly);
        o.w = bil(tl.w, tr.w, bl.w, br.w, lx, ly);

        const int cell = cell0 + s * kCellsPerStage + cq;
        *(float4*)&out[(size_t)cell * kC + w * 4] = o;     // 128-bit store

        if (s + 1 < kStagesPerBlock)
            __syncthreads();   // buffer (s&1) free for reuse at stage s+2
    }
}

extern "C" void kernel_launch(void* const* d_in, const int* in_sizes, int n_in,
                              void* d_out, int out_size, void* d_ws, size_t ws_size,
                              hipStream_t stream)
{
    (void)in_sizes; (void)n_in; (void)out_size; (void)d_ws; (void)ws_size;
    const float* f0    = (const float*)d_in[0];
    const float* f1    = (const float*)d_in[1];
    const float* f2    = (const float*)d_in[2];
    const float* f3    = (const float*)d_in[3];
    const float* props = (const float*)d_in[4];
    float* out         = (float*)d_out;

    hipLaunchKernelGGL(roi_align_async_kernel,
                       dim3(kBlocks), dim3(256), 0, stream,
                       f0, f1, f2, f3, props, out);
}